// VolumeRenderer_13245679140934
// MI455X (gfx1250) — compile-verified
//
#include <hip/hip_runtime.h>
#include <cstdint>

#define RAYS_PER_BLOCK 8
#define NPTS 128
#define EPSW 1e-10f
#define FARDELTA 1e10f

typedef __attribute__((address_space(1))) int gint_t;   // global int
typedef __attribute__((address_space(3))) int lint_t;   // LDS int

// ---- async global -> LDS, one b32 per lane ------------------------------
#if __has_builtin(__builtin_amdgcn_global_load_async_to_lds_b32)
#define ASYNC_VIA_BUILTIN 1
__device__ __forceinline__ void async_b32(const float* g, float* l) {
  __builtin_amdgcn_global_load_async_to_lds_b32((gint_t*)g, (lint_t*)l, 0, 0);
}
#else
#warning "global_load_async_to_lds_b32 builtin NOT available; using inline asm"
__device__ __forceinline__ void async_b32(const float* g, float* l) {
  unsigned lds_addr = (unsigned)(uintptr_t)(lint_t*)l;        // LDS byte offset
  unsigned long long ga = (unsigned long long)(uintptr_t)g;   // 64-bit global VA
  asm volatile("global_load_async_to_lds_b32 %0, %1, off"
               :: "v"(lds_addr), "v"(ga) : "memory");
}
#endif

#if __has_builtin(__builtin_amdgcn_s_wait_asynccnt)
#define WAIT_ASYNC(n) do { __builtin_amdgcn_s_wait_asynccnt(n); asm volatile("" ::: "memory"); } while (0)
#else
#define WAIT_ASYNC(n) asm volatile("s_wait_asynccnt %0" :: "n"(n) : "memory")
#endif

// ---- one wave32 per ray: shuffle-scan transmittance + weighted reduce ---
__global__ __launch_bounds__(256) void volrend_kernel(
    const float* __restrict__ density,
    const float* __restrict__ feature,
    const float* __restrict__ depth,
    float* __restrict__ out, int n_rays)
{
  __shared__ float sfeat[RAYS_PER_BLOCK * 2 * 96];  // 2 buffers x 32 pts x 3ch per wave
  const int lane = threadIdx.x & 31;
  const int wv   = threadIdx.x >> 5;
  const int ray  = blockIdx.x * RAYS_PER_BLOCK + wv;
  const int base = ray * NPTS;
  const float* fbase = feature + (size_t)base * 3;
  float* buf0 = &sfeat[(wv * 2 + 0) * 96];
  float* buf1 = &sfeat[(wv * 2 + 1) * 96];

  // prefetch chunk 0 feature tile into LDS (3 x b32 per lane = 96 floats)
  {
    const float* g = fbase + lane * 3;
    float* l = buf0 + lane * 3;
    async_b32(g + 0, l + 0); async_b32(g + 1, l + 1); async_b32(g + 2, l + 2);
  }
  // preload chunk 0 density/depth into registers
  int p0 = lane;
  float dcur  = density[base + p0];
  float zcur  = depth[base + p0];
  float znext = depth[base + (p0 < NPTS - 1 ? p0 + 1 : p0)];

  float carry = 1.0f;                 // running transmittance across chunks
  float r0 = 0.f, r1 = 0.f, r2 = 0.f, dep = 0.f;

  #pragma unroll
  for (int c = 0; c < 4; ++c) {
    float dnx = 0.f, znx = 0.f, znnx = 0.f;
    if (c < 3) {
      // issue async prefetch of chunk c+1 into the other buffer
      const float* g = fbase + (c + 1) * 96 + lane * 3;
      float* l = ((c + 1) & 1 ? buf1 : buf0) + lane * 3;
      async_b32(g + 0, l + 0); async_b32(g + 1, l + 1); async_b32(g + 2, l + 2);
      int pn = 32 * (c + 1) + lane;
      dnx  = density[base + pn];
      znx  = depth[base + pn];
      znnx = depth[base + (pn < NPTS - 1 ? pn + 1 : pn)];
      WAIT_ASYNC(3);                  // chunk c landed; c+1 may still be in flight
    } else {
      WAIT_ASYNC(0);                  // last chunk must be complete
    }
    const float* lb = (c & 1) ? buf1 : buf0;
    float f0 = lb[lane * 3 + 0];
    float f1 = lb[lane * 3 + 1];
    float f2 = lb[lane * 3 + 2];

    int pp = 32 * c + lane;
    float delta = (pp < NPTS - 1) ? (znext - zcur) : FARDELTA;
    float e     = __expf(-delta * dcur);
    float alpha = 1.0f - e;
    float u     = e + EPSW;           // 1 - alpha + eps

    // inclusive multiply-scan across 32 lanes (5 shuffle steps, wave32)
    float inc = u;
    #pragma unroll
    for (int s = 1; s < 32; s <<= 1) {
      float t = __shfl_up(inc, (unsigned)s, 32);
      if (lane >= s) inc *= t;
    }
    float excl = __shfl_up(inc, 1u, 32);
    if (lane == 0) excl = 1.0f;

    float w = carry * excl * alpha;   // weight = T * alpha
    r0  = fmaf(w, f0,   r0);
    r1  = fmaf(w, f1,   r1);
    r2  = fmaf(w, f2,   r2);
    dep = fmaf(w, zcur, dep);

    carry *= __shfl(inc, 31, 32);     // propagate transmittance to next chunk
    dcur = dnx; zcur = znx; znext = znnx;
  }

  // butterfly reduction over the 32 lanes
  #pragma unroll
  for (int s = 16; s >= 1; s >>= 1) {
    r0  += __shfl_xor(r0,  s, 32);
    r1  += __shfl_xor(r1,  s, 32);
    r2  += __shfl_xor(r2,  s, 32);
    dep += __shfl_xor(dep, s, 32);
  }
  if (lane == 0) {
    out[ray * 3 + 0] = r0;
    out[ray * 3 + 1] = r1;
    out[ray * 3 + 2] = r2;
    float* dout = out + (size_t)n_rays * 3;
    dout[ray * 3 + 0] = dep;
    dout[ray * 3 + 1] = dep;
    dout[ray * 3 + 2] = dep;
  }
}

extern "C" void kernel_launch(void* const* d_in, const int* in_sizes, int n_in,
                              void* d_out, int out_size, void* d_ws, size_t ws_size,
                              hipStream_t stream) {
  const float* density = (const float*)d_in[0];
  const float* feature = (const float*)d_in[1];
  const float* depth   = (const float*)d_in[2];
  float* out = (float*)d_out;
  int n_rays = in_sizes[0] / NPTS;                 // 65536
  dim3 grid(n_rays / RAYS_PER_BLOCK);              // 8192 blocks x 256 threads (8 waves)
  volrend_kernel<<<grid, 256, 0, stream>>>(density, feature, depth, out, n_rays);
}